// EfficientFB8NLLLoss_75917841924262
// MI455X (gfx1250) — compile-verified
//
#include <hip/hip_runtime.h>
#include <hip/hip_bf16.h>
#include <math.h>

// ---------------------------------------------------------------------------
// FB8 NLL loss, MI455X (gfx1250, wave32).
//  pass1: closed-form eigen setup per row (no 3x3 eigh needed), atomic-max es
//  pass2: one wave per row; lanes stripe the 2050 CET-FFT terms; streaming
//         complex log-sum-exp in registers; cross-lane sum via
//         V_WMMA_F32_16X16X4_F32 ones-matrix reduction (exact fp32)
//  pass3: deterministic final reduction -> mean
// ---------------------------------------------------------------------------

#define NB      4096
#define NFFT    1024
#define NTERMS  (2 * NFFT + 2)     // 2050
#define PI_F    3.14159265358979323846f

typedef __attribute__((ext_vector_type(2))) float v2f;
typedef __attribute__((ext_vector_type(8))) float v8f;

// workspace layout (floats)
#define WS_ES    0      // global max |gam|,|Lam| (nonneg float, atomic on bits)
#define WS_PART  16     // 512 block partial sums
#define WS_ROWS  1024   // 7 row arrays of NB: lam0,lam1,lam2,gam0,gam1,gam2,base

// ---- wave32 cross-lane sum of TWO scalars with one WMMA (ones-matrix trick).
// A(16x4): row i = [sr_i, si_i, sr_{i+16}, si_{i+16}]  (VGPR0=sr, VGPR1=si)
// B(4x16): cols 0..7 select K={0,2} (sr), cols 8..15 select K={1,3} (si)
// D[i][j<8] = sr_i + sr_{i+16},  D[i][j>=8] = si_i + si_{i+16}
__device__ __forceinline__ void wave_sum2_wmma(float sr, float si,
                                               float* SR, float* SI) {
  const int lane = threadIdx.x & 31;
  v2f a; a[0] = sr; a[1] = si;
  const float sel = ((lane & 15) < 8) ? 1.0f : 0.0f;
  v2f b; b[0] = sel; b[1] = 1.0f - sel;
  v8f acc = {};
  v8f d = __builtin_amdgcn_wmma_f32_16x16x4_f32(
      /*neg_a=*/false, a, /*neg_b=*/false, b,
      /*c_mod=*/(short)0, acc, /*reuse_a=*/false, /*reuse_b=*/false);
  // per-lane partial over 8 D rows
  float s = d[0] + d[1] + d[2] + d[3] + d[4] + d[5] + d[6] + d[7];
  // lanes 0-7:16..23 carry sr halves; lanes 8-15:24..31 carry si halves
  s += __shfl_xor(s, 16, 32);
  *SR = __shfl(s, 0, 32);
  *SI = __shfl(s, 8, 32);
}

__global__ void fb8_init(float* ws) {
  if (threadIdx.x == 0) ((unsigned*)ws)[WS_ES] = 0u;
}

// ---------------------------------------------------------------------------
// pass1: per-row closed-form setup. Bm's eigenbasis is {g1,g2,g3} (columns of
// G), eigenvalues {-t3, beta-t3, -beta*eta-t3}, gam = kappa*nu. quad and lin
// follow from dot products with x. Also feeds the global max 'es'.
// ---------------------------------------------------------------------------
__global__ __launch_bounds__(256) void fb8_pass1(
    const float* __restrict__ x,     const float* __restrict__ theta,
    const float* __restrict__ phi,   const float* __restrict__ psi,
    const float* __restrict__ kappa, const float* __restrict__ beta,
    const float* __restrict__ eta,   const float* __restrict__ alpha,
    const float* __restrict__ rho,   float* __restrict__ ws) {
  const int row = blockIdx.x * 256 + threadIdx.x;   // exact: 16*256 == NB

  __builtin_prefetch(theta + row, 0, 0);
  __builtin_prefetch(kappa + row, 0, 0);
  __builtin_prefetch(x + 3 * row, 0, 0);

  float st, ct, sp, cp, sps, cps, sa, ca, srho, crho;
  sincosf(theta[row], &st, &ct);
  sincosf(phi[row],   &sp, &cp);
  sincosf(psi[row],   &sps, &cps);
  sincosf(alpha[row], &sa, &ca);
  sincosf(rho[row],   &srho, &crho);
  const float kap = kappa[row], bet = beta[row], et = eta[row];

  // columns of G = H(theta,phi) @ K(psi)
  const float g1x = ct,        g1y = st * cp,                 g1z = st * sp;
  const float g2x = -st * cps, g2y = ct * cp * cps - sp * sps,
              g2z = ct * sp * cps + cp * sps;
  const float g3x = st * sps,  g3y = -ct * cp * sps - sp * cps,
              g3z = -ct * sp * sps + cp * cps;

  const float t3 = bet * (1.0f - et) * (1.0f / 3.0f);   // tr(Bm0)/3
  const float l0 = -t3;
  const float l1 = bet - t3;
  const float l2 = -bet * et - t3;

  const float gm0 = kap * ca;
  const float gm1 = kap * sa * crho;
  const float gm2 = kap * sa * srho;

  const float xx = x[3 * row + 0], xy = x[3 * row + 1], xz = x[3 * row + 2];
  const float d1 = g1x * xx + g1y * xy + g1z * xz;
  const float d2 = g2x * xx + g2y * xy + g2z * xz;
  const float d3 = g3x * xx + g3y * xy + g3z * xz;
  const float xn2 = xx * xx + xy * xy + xz * xz;

  const float quad = bet * (d2 * d2 - et * d3 * d3) - t3 * xn2;
  const float lin  = gm0 * d1 + gm1 * d2 + gm2 * d3;   // v . x

  float* rows = ws + WS_ROWS;
  rows[0 * NB + row] = l0;
  rows[1 * NB + row] = l1;
  rows[2 * NB + row] = l2;
  rows[3 * NB + row] = gm0;
  rows[4 * NB + row] = gm1;
  rows[5 * NB + row] = gm2;
  rows[6 * NB + row] = -quad - lin;

  float m = fmaxf(fmaxf(fabsf(l0), fabsf(l1)), fabsf(l2));
  m = fmaxf(m, fmaxf(fmaxf(fabsf(gm0), fabsf(gm1)), fabsf(gm2)));
  for (int off = 16; off; off >>= 1) m = fmaxf(m, __shfl_xor(m, off, 32));
  if ((threadIdx.x & 31) == 0)
    atomicMax((unsigned*)ws + WS_ES, __float_as_uint(m));  // nonneg: bit order == float order
}

// ---------------------------------------------------------------------------
// pass2: one wave per row; 65 strided iterations over 2050 terms; streaming
// complex log-sum-exp; WMMA lane reduction; 8 rows/block -> 1 partial/block.
// ---------------------------------------------------------------------------
__global__ __launch_bounds__(256) void fb8_pass2(const float* __restrict__ ws,
                                                 float* __restrict__ partial) {
  __shared__ float sdata[8];
  const int lane = threadIdx.x & 31;
  const int wave = threadIdx.x >> 5;
  const int row  = blockIdx.x * 8 + wave;

  const float* rows = ws + WS_ROWS;
  const float l0 = rows[0 * NB + row];
  const float l1 = rows[1 * NB + row];
  const float l2 = rows[2 * NB + row];
  const float g0 = rows[3 * NB + row];
  const float g1 = rows[4 * NB + row];
  const float g2 = rows[5 * NB + row];
  const float basev = rows[6 * NB + row];
  const float es = ws[WS_ES];

  const float cc = (es <= 10.0f) ? 25.0f
                 : (es <= 80.0f) ? 0.5f * es
                 : (es <= 250.0f) ? 0.6f * es : 0.7f * es;
  const float Lmax = fmaxf(l0, fmaxf(l1, l2));
  const float Ls0 = l0 - Lmax, Ls1 = l1 - Lmax, Ls2 = l2 - Lmax;
  const float mls  = fminf(Ls0, fminf(Ls1, Ls2));
  const float dfac = (es <= 80.0f) ? 0.25f : 0.1f;
  const float dd   = fmaxf(dfac * (cc + mls), 1e-9f);
  // h = sqrt(2*pi*d*(Od+Ou)/(Od^2*N)), Od=0.5, Ou=1.5, N=1024
  const float h  = sqrtf(2.0f * PI_F * dd * 2.0f / (0.25f * 1024.0f));
  const float p1 = sqrtf(1024.0f * h / 0.5f);
  const float p2 = sqrtf(0.5f * 1024.0f * h * 0.25f);
  const float ip1 = 1.0f / p1;

  const float dr0 = cc - Ls0, dr1 = cc - Ls1, dr2 = cc - Ls2;   // all > 0
  const float dq0 = dr0 * dr0, dq1 = dr1 * dr1, dq2 = dr2 * dr2;
  const float q0 = 0.25f * g0 * g0, q1 = 0.25f * g1 * g1, q2 = 0.25f * g2 * g2;
  const float id0 = 1.0f / dr0, id1 = 1.0f / dr1, id2 = 1.0f / dr2;

  float m = -1e30f, sr = 0.0f, si = 0.0f;   // streaming complex LSE state
  for (int it = 0; it < 65; ++it) {
    const int idx = lane + (it << 5);
    if (idx < NTERMS) {
      const float nf = (float)(idx - (NFFT + 1));
      const float nh = nf * h;
      const float nh2 = nh * nh;
      const float lw = __logf(0.5f * erfcf(fabsf(nh) * ip1 - p2));
      const float t0 = 1.0f / (dq0 + nh2);
      const float t1 = 1.0f / (dq1 + nh2);
      const float t2 = 1.0f / (dq2 + nh2);
      // lzr = log_w + sum_j( q_j*dr_j*t_j - 0.25*log(r2_j) )
      float la = lw + q0 * dr0 * t0 + q1 * dr1 * t1 + q2 * dr2 * t2
               - 0.25f * (__logf(dq0 + nh2) + __logf(dq1 + nh2) + __logf(dq2 + nh2));
      // lzi = nh - nh*sum_j(q_j*t_j) - 0.5*sum_j atan(nh/dr_j)
      const float ar = nh - nh * (q0 * t0 + q1 * t1 + q2 * t2)
                     - 0.5f * (atanf(nh * id0) + atanf(nh * id1) + atanf(nh * id2));
      la = fmaxf(la, -1e30f);                 // kills -inf from erfc underflow
      if (la > m) { const float sc = __expf(m - la); sr *= sc; si *= sc; m = la; }
      const float e = __expf(la - m);
      float cz, sz; sincosf(ar, &sz, &cz);
      sr = fmaf(e, cz, sr);
      si = fmaf(e, sz, si);
    }
  }

  // merge the 32 lane LSE states: shift to common max, then WMMA plain sum
  float M = m;
  for (int off = 16; off; off >>= 1) M = fmaxf(M, __shfl_xor(M, off, 32));
  const float sc = __expf(m - M);
  sr *= sc; si *= sc;
  float SR, SI;
  wave_sum2_wmma(sr, si, &SR, &SI);     // EXEC all-ones here (full wave, uniform)

  if (lane == 0) {
    const float Sabs = fmaxf(sqrtf(SR * SR + SI * SI), 1e-38f);
    const float logC = 0.5f * __logf(PI_F) + cc + __logf(h)
                     + __logf(Sabs) + M + Lmax;
    sdata[wave] = basev + logC;
  }
  __syncthreads();
  if (threadIdx.x == 0) {
    float s = 0.0f;
    for (int i = 0; i < 8; ++i) s += sdata[i];
    partial[blockIdx.x] = s;
  }
}

// ---------------------------------------------------------------------------
// pass3: deterministic reduction of 512 partials -> mean (no float atomics)
// ---------------------------------------------------------------------------
__global__ __launch_bounds__(256) void fb8_pass3(const float* __restrict__ ws,
                                                 float* __restrict__ out) {
  __shared__ float sdata[8];
  const int t = threadIdx.x;
  const float* part = ws + WS_PART;
  const float v = part[t] + part[t + 256];
  float SR, SI;
  wave_sum2_wmma(v, 0.0f, &SR, &SI);
  const int lane = t & 31, wave = t >> 5;
  if (lane == 0) sdata[wave] = SR;
  __syncthreads();
  if (t == 0) {
    float s = 0.0f;
    for (int i = 0; i < 8; ++i) s += sdata[i];
    out[0] = s * (1.0f / (float)NB);
  }
}

extern "C" void kernel_launch(void* const* d_in, const int* in_sizes, int n_in,
                              void* d_out, int out_size, void* d_ws, size_t ws_size,
                              hipStream_t stream) {
  const float* x     = (const float*)d_in[0];
  const float* theta = (const float*)d_in[1];
  const float* phi   = (const float*)d_in[2];
  const float* psi   = (const float*)d_in[3];
  const float* kappa = (const float*)d_in[4];
  const float* beta  = (const float*)d_in[5];
  const float* eta   = (const float*)d_in[6];
  const float* alpha = (const float*)d_in[7];
  const float* rho   = (const float*)d_in[8];
  float* wsf = (float*)d_ws;

  fb8_init <<<1,   64,  0, stream>>>(wsf);
  fb8_pass1<<<NB / 256, 256, 0, stream>>>(x, theta, phi, psi, kappa, beta, eta,
                                          alpha, rho, wsf);
  fb8_pass2<<<NB / 8, 256, 0, stream>>>(wsf, wsf + WS_PART);
  fb8_pass3<<<1, 256, 0, stream>>>(wsf, (float*)d_out);
}